// Conv2d_60009283059961
// MI455X (gfx1250) — compile-verified
//
#include <hip/hip_runtime.h>

// ---------------------------------------------------------------------------
// 7x7 stride-2 pad-3 single-channel conv on 8192x8192 fp32 -> 4096x4096 fp32.
//
// MI455X plan (compile-only reasoning):
//  * Memory floor: 268MB in + 67MB out @ 23.3TB/s  => ~14.4us. Needs ~114
//    TFLOP fp32 to keep up -> offload MACs to the matrix pipe via
//    V_WMMA_F32_16X16X4_F32 (keeps full fp32 precision vs the reference).
//  * Per input row r we compute all kernel-row responses
//        H(j,n) = sum_kw W(j,kw) * X(r, 2n+kw-3)
//    as D = A x B with A = weights (constant, rows j=0..6 AND duplicated in
//    rows 8..14 so both lane halves of D hold H), B = a strided slice of the
//    input row. K=7 padded to 8 -> 2 chained wmma_f32_16x16x4_f32.
//  * Stage 2: out(m,n) = sum_j H(2m+j-3, j, n). The lower lane half owns out
//    rows m0+e, the upper half m0+8+e; their rr-windows are disjoint, so the
//    gather is 112 branch-free masked v_fma_f32 (mask 0/1 per lane half) with
//    compile-time element indices -- no cndmask/select overhead, EXEC stays
//    uniform around the WMMAs (ISA requirement).
//  * Each wave (wave32) owns one 16x16 output tile; 8 waves/WG share a
//    37x268-float LDS input patch staged with GLOBAL_LOAD_ASYNC_TO_LDS_B128
//    (ASYNCcnt + s_wait_asynccnt) on interior blocks; border blocks use a
//    guarded zero-padding fill (matches conv zero padding).
// ---------------------------------------------------------------------------

typedef __attribute__((ext_vector_type(2))) float v2f;
typedef __attribute__((ext_vector_type(8))) float v8f;
typedef __attribute__((ext_vector_type(4))) int   v4i;

// Address-space-qualified v4i pointers for the async global->LDS DMA builtin
// (its signature per hipcc diagnostics: (v4i AS1*, v4i AS3*, int, int)).
typedef __attribute__((address_space(1))) v4i* as1_v4i_ptr;
typedef __attribute__((address_space(3))) v4i* as3_v4i_ptr;

#define NUMROW 8192
#define NUMCOL 8192
#define OUTH   4096
#define OUTW   4096

#define LDS_H  37     // input rows needed per 16-output-row band: 2*15+7 = 37
#define LDS_W  268    // dwords per LDS row (covers 262 needed cols, 16B-mult)

#if defined(__HIP_DEVICE_COMPILE__) && defined(__gfx1250__)
#if __has_builtin(__builtin_amdgcn_global_load_async_to_lds_b128)
#define HAVE_ASYNC_LDS 1
#endif
#endif

__global__ __launch_bounds__(256) void conv7x7_s2_wmma(
    const float* __restrict__ x,     // 8192*8192 input
    const float* __restrict__ w,     // 49 weights (7x7, row-major)
    const float* __restrict__ bias,  // 1
    float* __restrict__ out)         // 4096*4096 output
{
  __shared__ float smem[LDS_H * LDS_W];

  const int tid  = (int)threadIdx.x;
  const int wave = tid >> 5;         // wave32: 8 waves per 256-thread block
  const int lane = tid & 31;
  const int nloc = lane & 15;        // N coordinate (output column in tile)
  const int half = lane >> 4;        // 0: D rows 0-7, 1: D rows 8-15

  const int m0  = (int)blockIdx.y * 16;   // output row base of this block
  const int n0  = (int)blockIdx.x * 128;  // output col base (8 waves x 16)
  const int r0  = 2 * m0 - 3;             // first input row of the band
  const int c0a = 2 * n0 - 4;             // 16B-aligned first input col

  const bool interior = (r0 >= 0) && (r0 + LDS_H <= NUMROW) &&
                        (c0a >= 0) && (c0a + LDS_W <= NUMCOL);

  // -------- stage 0: stage the input patch into LDS --------
  if (interior) {
#if defined(HAVE_ASYNC_LDS)
    // 67 x 16B async DMA chunks per row, 37 rows; one instruction moves
    // 32 lanes x 16B = 512B. Addresses are 16B aligned (c0a multiple of 4
    // dwords, global row stride 8192 dwords, LDS stride 268 dwords = 67*16B).
    for (int t = tid; t < LDS_H * 67; t += 256) {
      const int rr = t / 67, seg = t % 67;
      const float* g = x + (size_t)(r0 + rr) * NUMCOL + (size_t)(c0a + seg * 4);
      float* l = &smem[rr * LDS_W + seg * 4];
      __builtin_amdgcn_global_load_async_to_lds_b128(
          (as1_v4i_ptr)g,   // const-drop + addrspacecast to global v4i*
          (as3_v4i_ptr)l,   // addrspacecast generic -> LDS v4i*
          0, 0);
    }
#if __has_builtin(__builtin_amdgcn_s_wait_asynccnt)
    __builtin_amdgcn_s_wait_asynccnt(0);
#else
    asm volatile("s_wait_asynccnt 0" ::: "memory");
#endif
#else
    for (int t = tid; t < LDS_H * LDS_W; t += 256) {
      const int rr = t / LDS_W, cc = t % LDS_W;
      smem[t] = x[(size_t)(r0 + rr) * NUMCOL + (size_t)(c0a + cc)];
    }
#endif
  } else {
    // Border: zero-fill out-of-range taps (== conv zero padding).
    for (int t = tid; t < LDS_H * LDS_W; t += 256) {
      const int rr = t / LDS_W, cc = t % LDS_W;
      const int gr = r0 + rr, gc = c0a + cc;
      float v = 0.0f;
      if (gr >= 0 && gr < NUMROW && gc >= 0 && gc < NUMCOL)
        v = x[(size_t)gr * NUMCOL + gc];
      smem[rr * LDS_W + cc] = v;
    }
  }
  __syncthreads();

  // -------- build the constant A operand (weights) --------
  // A 16x4 layout (ISA 7.12.2): lanes 0-15 -> M=lane, v0=K0, v1=K1;
  //                             lanes 16-31 -> M=lane-16, v0=K2, v1=K3.
  // A(j,kw)=W(j,kw) for j=0..6 AND A(8+j,kw)=W(j,kw) (duplicate) so that
  // D VGPR v carries H(j=v,n) in BOTH lane halves. Rows 7,15 and kw=7 are 0.
  const int mrow = lane & 15;
  const int jrow = (mrow < 7) ? mrow : ((mrow >= 8 && mrow < 15) ? mrow - 8 : -1);
  v2f a0, a1;
  {
    const int k0 = 2 * half;       // chunk0: kw = k0, k0+1   (kw 0..3)
    const int k1 = 4 + 2 * half;   // chunk1: kw = k1, k1+1   (kw 4..7)
    a0.x = (jrow >= 0)             ? w[jrow * 7 + k0]     : 0.0f;
    a0.y = (jrow >= 0)             ? w[jrow * 7 + k0 + 1] : 0.0f;
    a1.x = (jrow >= 0)             ? w[jrow * 7 + k1]     : 0.0f;
    a1.y = (jrow >= 0 && k1 < 6)   ? w[jrow * 7 + k1 + 1] : 0.0f; // kw=7 pad
  }

  // Per-lane 0/1 masks: lower half owns out rows m0+e, upper m0+8+e.
  const float mlo = (half == 0) ? 1.0f : 0.0f;
  const float mhi = 1.0f - mlo;

  // -------- stage 1+2: per input row, WMMA then banded masked-FMA gather ----
  // B 4x16 layout assumed symmetric to A: lanes 0-15 -> N=lane, v0=K0,v1=K1;
  // lanes 16-31 -> N=lane-16, v0=K2,v1=K3. Per lane the two elements are
  // adjacent input columns, LDS col = 32*wave + 2*n + 1 + 2*half (+4 chunk1).
  const int bcol = 32 * wave + 2 * nloc + 1 + 2 * half;

  v8f acc = (v8f)0.0f;
#pragma unroll
  for (int rr = 0; rr < LDS_H; ++rr) {
    const float* s = &smem[rr * LDS_W + bcol];
    v2f b0, b1;
    b0.x = s[0]; b0.y = s[1];
    b1.x = s[4]; b1.y = s[5];

    v8f h = (v8f)0.0f;
    h = __builtin_amdgcn_wmma_f32_16x16x4_f32(false, a0, false, b0,
                                              (short)0, h, false, false);
    h = __builtin_amdgcn_wmma_f32_16x16x4_f32(false, a1, false, b1,
                                              (short)0, h, false, false);

    // out(m0+ml, .) needs H from input row rr = 2*ml + j, j in [0,7).
    // Element e: lower half accumulates when j1 = rr-2e   in [0,7),
    //            upper half accumulates when j2 = rr-2e-16 in [0,7).
    // The windows are disjoint in rr, so each useful contribution is exactly
    // one masked FMA with compile-time constant indices (112 total).
#pragma unroll
    for (int e = 0; e < 8; ++e) {
      const int j1 = rr - 2 * e;
      if (j1 >= 0 && j1 < 7) acc[e] = __builtin_fmaf(h[j1], mlo, acc[e]);
      const int j2 = rr - 2 * e - 16;
      if (j2 >= 0 && j2 < 7) acc[e] = __builtin_fmaf(h[j2], mhi, acc[e]);
    }
  }

  // -------- stage 3: bias + store 16x16 tile --------
  const float bv   = bias[0];
  const int   orow = m0 + 8 * half;
  const int   ocol = n0 + 16 * wave + nloc;
#pragma unroll
  for (int e = 0; e < 8; ++e)
    out[(size_t)(orow + e) * OUTW + ocol] = acc[e] + bv;
}

extern "C" void kernel_launch(void* const* d_in, const int* in_sizes, int n_in,
                              void* d_out, int out_size, void* d_ws, size_t ws_size,
                              hipStream_t stream) {
  (void)in_sizes; (void)n_in; (void)d_ws; (void)ws_size; (void)out_size;
  const float* x    = (const float*)d_in[0];  // enc_x  (8192*8192)
  const float* w    = (const float*)d_in[1];  // weight (49)
  const float* bias = (const float*)d_in[2];  // bias   (1)
  float* out = (float*)d_out;                 // 4096*4096

  dim3 grid(OUTW / 128, OUTH / 16);  // 32 x 256 blocks
  conv7x7_s2_wmma<<<grid, 256, 0, stream>>>(x, w, bias, out);
}